// BaseLSSFPN_67327907332198
// MI455X (gfx1250) — compile-verified
//
#include <hip/hip_runtime.h>
#include <hip/hip_bf16.h>

typedef __attribute__((ext_vector_type(2))) float v2f;
typedef __attribute__((ext_vector_type(8))) float v8f;

#define NCAM 6
#define CCH  80
#define DD   112
#define FH_  16
#define FW_  44
#define PIX  (FH_ * FW_)          // 704
#define NPIX (NCAM * PIX)         // 4224
#define NDHW (NCAM * DD * PIX)    // 473088
#define OUTHW (128 * 128)         // 16384
#define OUT_ELEMS (CCH * OUTHW)   // 1310720

// ---------------- 4x4 inverse (Gauss-Jordan, per-thread) ----------------
__device__ inline void inv4(const float* m, float* out) {
    float a[4][8];
    for (int i = 0; i < 4; ++i) {
        for (int j = 0; j < 4; ++j) {
            a[i][j]     = m[i * 4 + j];
            a[i][j + 4] = (i == j) ? 1.0f : 0.0f;
        }
    }
    for (int col = 0; col < 4; ++col) {
        int piv = col;
        float best = fabsf(a[col][col]);
        for (int r = col + 1; r < 4; ++r) {
            float v = fabsf(a[r][col]);
            if (v > best) { best = v; piv = r; }
        }
        if (piv != col) {
            for (int j = 0; j < 8; ++j) {
                float t = a[col][j]; a[col][j] = a[piv][j]; a[piv][j] = t;
            }
        }
        float d = 1.0f / a[col][col];
        for (int j = 0; j < 8; ++j) a[col][j] *= d;
        for (int r = 0; r < 4; ++r) {
            if (r == col) continue;
            float f = a[r][col];
            for (int j = 0; j < 8; ++j) a[r][j] -= f * a[col][j];
        }
    }
    for (int i = 0; i < 4; ++i)
        for (int j = 0; j < 4; ++j)
            out[i * 4 + j] = a[i][j + 4];
}

// mats[n*32 + 0..15]  = inv(ida[n])
// mats[n*32 + 16..31] = bda @ s2e[n] @ inv(intr[n])
__global__ void k_mats(const float* __restrict__ s2e, const float* __restrict__ intr,
                       const float* __restrict__ ida, const float* __restrict__ bda,
                       float* __restrict__ mats) {
    int n = threadIdx.x;
    if (n >= NCAM) return;
    float invIda[16], invIntr[16], T[16], Mm[16];
    inv4(ida + n * 16, invIda);
    inv4(intr + n * 16, invIntr);
    for (int i = 0; i < 4; ++i)
        for (int j = 0; j < 4; ++j) {
            float s = 0.0f;
            for (int k = 0; k < 4; ++k) s += s2e[n * 16 + i * 4 + k] * invIntr[k * 4 + j];
            T[i * 4 + j] = s;
        }
    for (int i = 0; i < 4; ++i)
        for (int j = 0; j < 4; ++j) {
            float s = 0.0f;
            for (int k = 0; k < 4; ++k) s += bda[i * 4 + k] * T[k * 4 + j];
            Mm[i * 4 + j] = s;
        }
    for (int i = 0; i < 16; ++i) {
        mats[n * 32 + i]      = invIda[i];
        mats[n * 32 + 16 + i] = Mm[i];
    }
}

// ---------------- per-pixel depth softmax (unnormalized exp + 1/sum) ----------------
__global__ void k_softmax(const float* __restrict__ logits, float* __restrict__ expd,
                          float* __restrict__ invsum) {
    int p = blockIdx.x * blockDim.x + threadIdx.x;
    if (p >= NPIX) return;
    int n  = p / PIX;
    int hw = p - n * PIX;
    const float* lp = logits + n * DD * PIX + hw;
    float mx = -3.4e38f;
    for (int d = 0; d < DD; ++d) mx = fmaxf(mx, lp[d * PIX]);
    float s = 0.0f;
    float* ep = expd + n * DD * PIX + hw;
    for (int d = 0; d < DD; ++d) {
        float e = __expf(lp[d * PIX] - mx);
        ep[d * PIX] = e;
        s += e;
    }
    invsum[p] = 1.0f / s;
}

// ---------------- per-point voxel index ----------------
__global__ void k_geom(const float* __restrict__ mats, int* __restrict__ vidx) {
    int t = blockIdx.x * blockDim.x + threadIdx.x;
    if (t >= NDHW) return;
    int n   = t / (DD * PIX);
    int rem = t - n * (DD * PIX);
    int d   = rem / PIX;
    int hw  = rem - d * PIX;
    int h = hw / FW_;
    int w = hw - h * FW_;

    const float* Ai = mats + n * 32;
    const float* M  = mats + n * 32 + 16;

    float x   = (float)w * (703.0f / 43.0f);   // linspace(0, 703, 44)
    float y   = (float)h * 17.0f;              // linspace(0, 255, 16)
    float dep = 2.0f + 0.5f * (float)d;

    float q0 = Ai[0]  * x + Ai[1]  * y + Ai[2]  * dep + Ai[3];
    float q1 = Ai[4]  * x + Ai[5]  * y + Ai[6]  * dep + Ai[7];
    float q2 = Ai[8]  * x + Ai[9]  * y + Ai[10] * dep + Ai[11];
    float q3 = Ai[12] * x + Ai[13] * y + Ai[14] * dep + Ai[15];

    float u0 = q0 * q2, u1 = q1 * q2, u2 = q2, u3 = q3;

    float p0 = M[0] * u0 + M[1] * u1 + M[2]  * u2 + M[3]  * u3;
    float p1 = M[4] * u0 + M[5] * u1 + M[6]  * u2 + M[7]  * u3;
    float p2 = M[8] * u0 + M[9] * u1 + M[10] * u2 + M[11] * u3;

    const float bx0 = -50.8f - 0.8f * 0.5f;   // same f32 arithmetic as reference
    const float bx2 = -1.0f  - 8.0f * 0.5f;
    int vx = (int)floorf((p0 - bx0) / 0.8f);
    int vy = (int)floorf((p1 - bx0) / 0.8f);
    int vz = (int)floorf((p2 - bx2) / 8.0f);

    bool ok = (vx >= 0) & (vx < 128) & (vy >= 0) & (vy < 128) & (vz == 0);
    vidx[t] = ok ? (vx * 128 + vy) : -1;
}

__global__ void k_zero(float* __restrict__ out, int nel) {
    int i = blockIdx.x * blockDim.x + threadIdx.x;
    if (i < nel) out[i] = 0.0f;
}

// ---------------- main: WMMA outer product + atomic scatter ----------------
// One wave per (pixel, 16-channel tile). 7 depth tiles of 16 per pixel.
// All 112 exp-depth values and 112 voxel indices for the pixel are loaded once
// into 4 registers each (lane l holds m = l, l+32, l+64, l+96) and redistributed
// per depth tile with ds_bpermute shuffles:
//   tile dt covers m = dt*16 .. dt*16+15, entirely inside register dt/2,
//   at source lane ((dt&1)<<4) + j.
// A (16x4 f32): A[m,0] = exp_depth[m], K=1..3 zero
// B (4x16 f32): B[0,n] = ctx[n]*invsum, rows 1..3 zero
// D (16x16 f32): lane l, VGPR r -> (M = r + 8*(l>=16), N = l&15)
__global__ void k_main(const float* __restrict__ ctx, const float* __restrict__ expd,
                       const float* __restrict__ invsum, const int* __restrict__ vidx,
                       float* __restrict__ out) {
    int wave = blockIdx.x * (blockDim.x >> 5) + (threadIdx.x >> 5);
    int lane = threadIdx.x & 31;
    int pid = wave / 5;           // pixel id in [0, NPIX)
    int ct  = wave - pid * 5;     // channel tile in [0, 5)
    if (pid >= NPIX) return;      // wave-uniform

    int n  = pid / PIX;
    int hw = pid - n * PIX;
    int nlo = lane & 15;
    int half8 = (lane >> 4) << 3;            // 0 for lanes 0-15, 8 for 16-31

    float is = invsum[pid];
    v2f b; b.x = 0.0f; b.y = 0.0f;
    if (lane < 16)
        b.x = ctx[(n * CCH + ct * 16 + lane) * PIX + hw] * is;

    const float* ep = expd + n * DD * PIX + hw;
    const int*   vp = vidx + n * DD * PIX + hw;
    float* outc = out + (ct * 16 + nlo) * OUTHW;

    // Batched wave-resident loads: lane l holds element m = l + 32*k.
    float ereg[4];
    int   vreg[4];
    ereg[0] = ep[lane * PIX];
    ereg[1] = ep[(lane + 32) * PIX];
    ereg[2] = ep[(lane + 64) * PIX];
    ereg[3] = (lane < 16) ? ep[(lane + 96) * PIX] : 0.0f;   // m=96..111 only
    vreg[0] = vp[lane * PIX];
    vreg[1] = vp[(lane + 32) * PIX];
    vreg[2] = vp[(lane + 64) * PIX];
    vreg[3] = (lane < 16) ? vp[(lane + 96) * PIX] : -1;

#pragma unroll
    for (int dt = 0; dt < 7; ++dt) {
        const int reg  = dt >> 1;
        const int lbase = (dt & 1) << 4;     // lane offset of this tile inside reg

        // A operand: shuffle with full EXEC, then mask to lanes 0-15.
        float av = __shfl(ereg[reg], lbase + nlo, 32);
        v2f a; a.x = (lane < 16) ? av : 0.0f; a.y = 0.0f;

        v8f acc = {0.0f, 0.0f, 0.0f, 0.0f, 0.0f, 0.0f, 0.0f, 0.0f};
        acc = __builtin_amdgcn_wmma_f32_16x16x4_f32(
            /*neg_a=*/false, a, /*neg_b=*/false, b,
            /*c_mod=*/(short)0, acc, /*reuse_a=*/false, /*reuse_b=*/false);

#pragma unroll
        for (int r = 0; r < 8; ++r) {
            int vi = __shfl(vreg[reg], lbase + r + half8, 32);
            if (vi >= 0)
                atomicAdd(outc + vi, acc[r]);
        }
    }
}

extern "C" void kernel_launch(void* const* d_in, const int* in_sizes, int n_in,
                              void* d_out, int out_size, void* d_ws, size_t ws_size,
                              hipStream_t stream) {
    const float* ctx    = (const float*)d_in[0];
    const float* logits = (const float*)d_in[1];
    const float* s2e    = (const float*)d_in[2];
    const float* intr   = (const float*)d_in[3];
    const float* ida    = (const float*)d_in[4];
    const float* bda    = (const float*)d_in[5];
    float* out = (float*)d_out;

    // workspace layout (floats): [0,192) mats | [256,4480) invsum | [4480,477568) expd | then vidx (ints)
    float* mats   = (float*)d_ws;
    float* invsum = mats + 256;
    float* expd   = invsum + NPIX;
    int*   vidx   = (int*)(expd + NDHW);

    k_mats<<<1, 32, 0, stream>>>(s2e, intr, ida, bda, mats);
    k_softmax<<<(NPIX + 255) / 256, 256, 0, stream>>>(logits, expd, invsum);
    k_geom<<<(NDHW + 255) / 256, 256, 0, stream>>>(mats, vidx);
    k_zero<<<(OUT_ELEMS + 255) / 256, 256, 0, stream>>>(out, OUT_ELEMS);

    const int totalWaves = NPIX * 5;                 // 21120
    k_main<<<totalWaves / 8, 256, 0, stream>>>(ctx, expd, invsum, vidx, out);
}